// ExpansionContrastModule_7352984011472
// MI455X (gfx1250) — compile-verified
//
#include <hip/hip_runtime.h>

typedef __attribute__((ext_vector_type(2))) float v2f;
typedef __attribute__((ext_vector_type(8))) float v8f;

#define Bn   4
#define ICn  64
#define Cc   16
#define Hd   256
#define Wd   256
#define HW   65536              // Hd*Wd
#define NPIX (Bn * HW)          // 262144 pixels total
#define NTILE (NPIX / 16)       // 16384 wave-tiles of 16 pixels

// Interleaved (NHWC) intermediates: element (pix, c) at pix*16 + c.
// One pixel's 16 channels = 64 B = one cacheline.

__device__ __forceinline__ float4 ld4(const float* p) {
    return *reinterpret_cast<const float4*>(p);
}
__device__ __forceinline__ void st4(float* p, float4 v) {
    *reinterpret_cast<float4*>(p) = v;
}

// ---------------------------------------------------------------------------
// Kernel 1: x0 = 1x1 conv(cen, w_in) + b_in      (64 -> 16 channels)
// D(16x16) = W(16x64) * X(64x16pix) via 16 chained K=4 WMMAs per tile.
// cen stays planar (read once from HBM); x0 written NHWC.
// ---------------------------------------------------------------------------
__global__ __launch_bounds__(256) void k_in_conv(
    const float* __restrict__ cen, const float* __restrict__ w_in,
    const float* __restrict__ b_in, float* __restrict__ x0)
{
    __shared__ float wlds[Cc * ICn];                 // 4 KB, [c_out][c_in]
    for (int i = threadIdx.x; i < Cc * ICn; i += blockDim.x) wlds[i] = w_in[i];
    __syncthreads();

    const int wave = blockIdx.x * (blockDim.x >> 5) + (threadIdx.x >> 5);
    const int lane = threadIdx.x & 31;
    const int half = lane >> 4;
    const int lid  = lane & 15;
    const int b     = wave / (HW / 16);
    const int pbase = (wave % (HW / 16)) * 16;
    const float* src = cen + (size_t)b * ICn * HW;

    v8f acc = {};
#pragma unroll
    for (int kb = 0; kb < 16; ++kb) {
        const int k0 = 4 * kb + 2 * half;
        v2f a, bf;
        a.x  = wlds[lid * ICn + k0];
        a.y  = wlds[lid * ICn + k0 + 1];
        bf.x = src[(size_t)k0 * HW + pbase + lid];
        bf.y = src[(size_t)(k0 + 1) * HW + pbase + lid];
        acc = __builtin_amdgcn_wmma_f32_16x16x4_f32(false, a, false, bf,
                                                    (short)0, acc, false, false);
    }
    // lane holds rows m = half*8 .. half*8+7 for pixel (pbase+lid): contiguous NHWC
    const float4 bias0 = ld4(b_in + half * 8);
    const float4 bias1 = ld4(b_in + half * 8 + 4);
    float* dst = x0 + ((size_t)b * HW + pbase + lid) * Cc + half * 8;
    st4(dst,     make_float4(acc[0] + bias0.x, acc[1] + bias0.y,
                             acc[2] + bias0.z, acc[3] + bias0.w));
    st4(dst + 4, make_float4(acc[4] + bias1.x, acc[5] + bias1.y,
                             acc[6] + bias1.z, acc[7] + bias1.w));
}

// ---------------------------------------------------------------------------
// Kernel 2: xs = conv_SxS(x0, ws) + bs           (16 -> 16 channels, NHWC)
// Weights re-laid in LDS as [tap][c_out][c_in] -> contiguous b64 A-fragments.
// Each wave accumulates 2 tiles (32 pixels) to amortize A-fragment loads.
// ---------------------------------------------------------------------------
template <int S>
__global__ __launch_bounds__(256) void k_branch_conv(
    const float* __restrict__ x0, const float* __restrict__ ws,
    const float* __restrict__ bs, float* __restrict__ xs)
{
    __shared__ float wlds[S * S * Cc * Cc];          // up to 50 KB (S=7)
    for (int i = threadIdx.x; i < S * S * Cc * Cc; i += blockDim.x) {
        const int tap = i >> 8;                      // i / 256
        const int rem = i & 255;
        const int co  = rem >> 4;
        const int ci  = rem & 15;
        wlds[i] = ws[(co * Cc + ci) * (S * S) + tap];
    }
    __syncthreads();

    const int wave = blockIdx.x * (blockDim.x >> 5) + (threadIdx.x >> 5);
    const int lane = threadIdx.x & 31;
    const int half = lane >> 4;
    const int lid  = lane & 15;
    const int base = wave * 32;                      // 2 tiles per wave
    const int b    = base / HW;
    const int p0   = base % HW;
    const int y    = p0 >> 8;
    const int xb   = p0 & (Wd - 1);
    const int pad  = S >> 1;
    const float* src = x0 + (size_t)b * HW * Cc;

    v8f acc0 = {}, acc1 = {};
#pragma unroll
    for (int ky = 0; ky < S; ++ky) {
        const int yy  = y + ky - pad;
        const bool yok = (unsigned)yy < (unsigned)Hd;
#pragma unroll
        for (int kx = 0; kx < S; ++kx) {
            const float* wtap = wlds + ((ky * S + kx) * Cc + lid) * Cc;
            const int xx0 = xb + lid + kx - pad;         // tile 0 column
            const int xx1 = xx0 + 16;                    // tile 1 column
            const bool ok0 = yok && ((unsigned)xx0 < (unsigned)Wd);
            const bool ok1 = yok && ((unsigned)xx1 < (unsigned)Wd);
            const float* px0 = src + ((size_t)yy * Wd + xx0) * Cc;
            const float* px1 = src + ((size_t)yy * Wd + xx1) * Cc;
#pragma unroll
            for (int kb = 0; kb < 4; ++kb) {
                const int k0 = 4 * kb + 2 * half;
                v2f a, bf0, bf1;
                a.x = wtap[k0];
                a.y = wtap[k0 + 1];
                bf0.x = ok0 ? px0[k0]     : 0.0f;
                bf0.y = ok0 ? px0[k0 + 1] : 0.0f;
                bf1.x = ok1 ? px1[k0]     : 0.0f;
                bf1.y = ok1 ? px1[k0 + 1] : 0.0f;
                acc0 = __builtin_amdgcn_wmma_f32_16x16x4_f32(false, a, false, bf0,
                                                             (short)0, acc0, false, false);
                acc1 = __builtin_amdgcn_wmma_f32_16x16x4_f32(false, a, false, bf1,
                                                             (short)0, acc1, false, false);
            }
        }
    }
    const float4 bias0 = ld4(bs + half * 8);
    const float4 bias1 = ld4(bs + half * 8 + 4);
    float* dst0 = xs + ((size_t)b * HW + p0 + lid) * Cc + half * 8;
    float* dst1 = dst0 + 16 * Cc;
    st4(dst0,     make_float4(acc0[0] + bias0.x, acc0[1] + bias0.y,
                              acc0[2] + bias0.z, acc0[3] + bias0.w));
    st4(dst0 + 4, make_float4(acc0[4] + bias1.x, acc0[5] + bias1.y,
                              acc0[6] + bias1.z, acc0[7] + bias1.w));
    st4(dst1,     make_float4(acc1[0] + bias0.x, acc1[1] + bias0.y,
                              acc1[2] + bias0.z, acc1[3] + bias0.w));
    st4(dst1 + 4, make_float4(acc1[4] + bias1.x, acc1[5] + bias1.y,
                              acc1[6] + bias1.z, acc1[7] + bias1.w));
}

// ---------------------------------------------------------------------------
// Kernel 3: 8 dilated differences -> pair-sum grouped 1x1 -> relu -> sort8
//           -> grouped 1x1 -> running max / sum across the 4 branches.
// NHWC: process 4 channels per float4 group; 9 b128 loads per group.
// ---------------------------------------------------------------------------
__device__ __forceinline__ void cswap(float& a, float& b) {
    const float lo = fminf(a, b), hi = fmaxf(a, b);
    a = lo; b = hi;
}

template <int S, int FIRST>
__global__ __launch_bounds__(256) void k_branch_point(
    const float* __restrict__ xs,
    const float* __restrict__ w1, const float* __restrict__ b1,
    const float* __restrict__ w2, const float* __restrict__ b2,
    float* __restrict__ bmax, float* __restrict__ bsum)
{
    const int gid = blockIdx.x * blockDim.x + threadIdx.x;   // pixel id
    const int b = gid / HW;
    const int p = gid % HW;
    const int y = p >> 8;
    const int x = p & (Wd - 1);

    const int dy[8] = {-1, -1, -1, 0, 1, 1, 1, 0};
    const int dx[8] = {-1,  0,  1, 1, 1, 0, -1, -1};

    const float* pix = xs + ((size_t)b * HW + p) * Cc;

#pragma unroll
    for (int cg = 0; cg < 4; ++cg) {                 // channel group of 4
        const float4 ctr = ld4(pix + 4 * cg);
        float4 t[8];
#pragma unroll
        for (int m = 0; m < 8; ++m) {
            const int yy = y + dy[m] * S;
            const int xx = x + dx[m] * S;
            const bool ok = ((unsigned)yy < (unsigned)Hd) && ((unsigned)xx < (unsigned)Wd);
            float4 nb = ok ? ld4(xs + (((size_t)b * HW + yy * Wd + xx) * Cc + 4 * cg))
                           : make_float4(0.f, 0.f, 0.f, 0.f);
            t[m] = make_float4(ctr.x - nb.x, ctr.y - nb.y, ctr.z - nb.z, ctr.w - nb.w);
        }

        float out4[4];
#pragma unroll
        for (int cl = 0; cl < 4; ++cl) {             // channel within group
            const int c = 4 * cg + cl;
            float tc[8];
#pragma unroll
            for (int m = 0; m < 8; ++m)
                tc[m] = (cl == 0) ? t[m].x : (cl == 1) ? t[m].y : (cl == 2) ? t[m].z : t[m].w;

            const float4 W1 = ld4(w1 + c * 4);
            const float4 W2a = ld4(w2 + c * 8);
            const float4 W2b = ld4(w2 + c * 8 + 4);
            const float B1 = b1[c], B2 = b2[c];

            float o[8];
#pragma unroll
            for (int m = 0; m < 8; ++m) {
                float s = B1;
                s = fmaf(W1.x, tc[(m + 1) & 7] + tc[(m + 7) & 7], s);
                s = fmaf(W1.y, tc[(m + 2) & 7] + tc[(m + 6) & 7], s);
                s = fmaf(W1.z, tc[(m + 3) & 7] + tc[(m + 5) & 7], s);
                s = fmaf(W1.w, tc[(m + 4) & 7] + tc[(m + 4) & 7], s);
                o[m] = fmaxf(s * tc[m], 0.0f);
            }
            // Batcher odd-even mergesort, 19 comparators, ascending
            cswap(o[0], o[1]); cswap(o[2], o[3]); cswap(o[4], o[5]); cswap(o[6], o[7]);
            cswap(o[0], o[2]); cswap(o[1], o[3]); cswap(o[4], o[6]); cswap(o[5], o[7]);
            cswap(o[1], o[2]); cswap(o[5], o[6]);
            cswap(o[0], o[4]); cswap(o[1], o[5]); cswap(o[2], o[6]); cswap(o[3], o[7]);
            cswap(o[2], o[4]); cswap(o[3], o[5]);
            cswap(o[1], o[2]); cswap(o[3], o[4]); cswap(o[5], o[6]);

            float r = B2;
            r = fmaf(W2a.x, o[0], r); r = fmaf(W2a.y, o[1], r);
            r = fmaf(W2a.z, o[2], r); r = fmaf(W2a.w, o[3], r);
            r = fmaf(W2b.x, o[4], r); r = fmaf(W2b.y, o[5], r);
            r = fmaf(W2b.z, o[6], r); r = fmaf(W2b.w, o[7], r);
            out4[cl] = r;
        }

        float* pm = bmax + ((size_t)b * HW + p) * Cc + 4 * cg;
        float* psum = bsum + ((size_t)b * HW + p) * Cc + 4 * cg;
        const float4 nv = make_float4(out4[0], out4[1], out4[2], out4[3]);
        if (FIRST) {
            st4(pm, nv);
            st4(psum, nv);
        } else {
            const float4 om = ld4(pm);
            const float4 os = ld4(psum);
            st4(pm, make_float4(fmaxf(om.x, nv.x), fmaxf(om.y, nv.y),
                                fmaxf(om.z, nv.z), fmaxf(om.w, nv.w)));
            st4(psum, make_float4(os.x + nv.x, os.y + nv.y,
                                  os.z + nv.z, os.w + nv.w));
        }
    }
}

// ---------------------------------------------------------------------------
// Kernel 4: comb = max + mean ; att = sigmoid(<comb, w_out>) ; out = att*cen
// ---------------------------------------------------------------------------
__global__ __launch_bounds__(256) void k_final(
    const float* __restrict__ cen, const float* __restrict__ bmax,
    const float* __restrict__ bsum, const float* __restrict__ w_out,
    float* __restrict__ out)
{
    const int gid = blockIdx.x * blockDim.x + threadIdx.x;   // pixel id
    const int b = gid / HW;
    const int p = gid % HW;

    const size_t pix = ((size_t)b * HW + p) * Cc;
    float acc = 0.0f;
#pragma unroll
    for (int c4 = 0; c4 < 4; ++c4) {
        const float4 mx = ld4(bmax + pix + 4 * c4);
        const float4 sm = ld4(bsum + pix + 4 * c4);
        const float4 wv = ld4(w_out + 4 * c4);
        acc = fmaf(mx.x + 0.25f * sm.x, wv.x, acc);
        acc = fmaf(mx.y + 0.25f * sm.y, wv.y, acc);
        acc = fmaf(mx.z + 0.25f * sm.z, wv.z, acc);
        acc = fmaf(mx.w + 0.25f * sm.w, wv.w, acc);
    }
    const float att = 1.0f / (1.0f + __expf(-acc));
    const size_t base = (size_t)b * ICn * HW + p;
#pragma unroll 8
    for (int ic = 0; ic < ICn; ++ic)
        out[base + (size_t)ic * HW] = att * cen[base + (size_t)ic * HW];
}

// ---------------------------------------------------------------------------
extern "C" void kernel_launch(void* const* d_in, const int* in_sizes, int n_in,
                              void* d_out, int out_size, void* d_ws, size_t ws_size,
                              hipStream_t stream)
{
    const float* cen   = (const float*)d_in[0];
    // d_in[1] = mas (unused by reference)
    const float* w_in  = (const float*)d_in[2];
    const float* b_in  = (const float*)d_in[3];
    const float* wsk[4] = {(const float*)d_in[4], (const float*)d_in[6],
                           (const float*)d_in[8], (const float*)d_in[10]};
    const float* bsk[4] = {(const float*)d_in[5], (const float*)d_in[7],
                           (const float*)d_in[9], (const float*)d_in[11]};
    const float* w1    = (const float*)d_in[12];
    const float* b1    = (const float*)d_in[13];
    const float* w2    = (const float*)d_in[14];
    const float* b2    = (const float*)d_in[15];
    const float* w_out = (const float*)d_in[16];
    float* out = (float*)d_out;

    const size_t plane = (size_t)Bn * HW * Cc;       // 4.19M floats = 16.7 MB
    float* x0   = (float*)d_ws;                      // NHWC
    float* xs   = x0 + plane;                        // NHWC
    float* bmax = xs + plane;                        // NHWC
    float* bsum = bmax + plane;                      // NHWC

    const dim3 blk(256);
    const dim3 gTile(NTILE / 8);                     // 1 tile/wave, 8 waves/block
    const dim3 gTile2(NTILE / 16);                   // 2 tiles/wave
    const dim3 gPix(NPIX / 256);

    k_in_conv<<<gTile, blk, 0, stream>>>(cen, w_in, b_in, x0);

    k_branch_conv<1><<<gTile2, blk, 0, stream>>>(x0, wsk[0], bsk[0], xs);
    k_branch_point<1, 1><<<gPix, blk, 0, stream>>>(xs, w1, b1, w2, b2, bmax, bsum);

    k_branch_conv<3><<<gTile2, blk, 0, stream>>>(x0, wsk[1], bsk[1], xs);
    k_branch_point<3, 0><<<gPix, blk, 0, stream>>>(xs, w1, b1, w2, b2, bmax, bsum);

    k_branch_conv<5><<<gTile2, blk, 0, stream>>>(x0, wsk[2], bsk[2], xs);
    k_branch_point<5, 0><<<gPix, blk, 0, stream>>>(xs, w1, b1, w2, b2, bmax, bsum);

    k_branch_conv<7><<<gTile2, blk, 0, stream>>>(x0, wsk[3], bsk[3], xs);
    k_branch_point<7, 0><<<gPix, blk, 0, stream>>>(xs, w1, b1, w2, b2, bmax, bsum);

    k_final<<<gPix, blk, 0, stream>>>(cen, bmax, bsum, w_out, out);
}